// GraphConvNetwork_1597727834802
// MI455X (gfx1250) — compile-verified
//
#include <hip/hip_runtime.h>

// GraphConvNetwork on MI455X (gfx1250, wave32)
// layer: agg = scatter_add(edge_w * h[src] -> dst); out = agg@w_rel + b + h@w_root [+relu]
// fp32 WMMA (V_WMMA_F32_16X16X4_F32) keeps reference precision; op is L2/atomic bound.

typedef __attribute__((ext_vector_type(2))) float v2f;
typedef __attribute__((ext_vector_type(8))) float v8f;

#define N_NODES 50000
#define N_EDGES 800000
#define DIM 128
#define STRIPES (N_NODES / 16)   // 3125 exact

// ---------------- zero the aggregation buffer (float4 stores) ----------------
__global__ __launch_bounds__(256) void zero_f4(float4* __restrict__ p, int n4) {
    int i = blockIdx.x * blockDim.x + threadIdx.x;
    if (i < n4) p[i] = make_float4(0.f, 0.f, 0.f, 0.f);
}

// ---------------- edge scatter: one wave per edge, one float4 per lane -------
__global__ __launch_bounds__(256) void scatter_edges(
    const float* __restrict__ h,
    const int*   __restrict__ src,
    const int*   __restrict__ dst,
    const float* __restrict__ ew,
    float*       __restrict__ agg)
{
    const int gtid = blockIdx.x * blockDim.x + threadIdx.x;
    const int e    = gtid >> 5;        // wave-uniform edge id
    const int lane = gtid & 31;        // lane owns features [lane*4, lane*4+3]
    if (e >= N_EDGES) return;

    const int   s = src[e];
    const int   d = dst[e];
    const float w = ew[e];

    const float4 v = ((const float4*)(h + (size_t)s * DIM))[lane];
    float* o = agg + (size_t)d * DIM + lane * 4;
    unsafeAtomicAdd(o + 0, v.x * w);
    unsafeAtomicAdd(o + 1, v.y * w);
    unsafeAtomicAdd(o + 2, v.z * w);
    unsafeAtomicAdd(o + 3, v.w * w);
}

// ---------------- fused dual-GEMM + bias + relu via f32 WMMA -----------------
// block = 256 threads = 8 waves; block handles one 16-row stripe,
// wave w computes the 16x16 tile at columns [16w, 16w+15].
// A frag (16x4 f32): lane 0-15 -> M=lane, K=k..k+1 ; lane 16-31 -> M=lane-16, K=k+2..k+3
// B frag (4x16 f32): lane 0-15 -> N=lane, K=k..k+1 ; lane 16-31 -> N=lane-16, K=k+2..k+3
// C/D (16x16 f32): VGPR g, lane<16 -> (M=g,   N=lane); lane>=16 -> (M=g+8, N=lane-16)
__global__ __launch_bounds__(256) void gemm_stripe(
    const float* __restrict__ agg,
    const float* __restrict__ h,
    const float* __restrict__ w_rel,
    const float* __restrict__ b_rel,
    const float* __restrict__ w_root,
    float*       __restrict__ out,
    int do_relu)
{
    const int stripe = blockIdx.x;          // 0..STRIPES-1
    const int wave   = threadIdx.x >> 5;    // 0..7  -> column tile
    const int lane   = threadIdx.x & 31;

    const int row0 = stripe * 16;
    const int col0 = wave * 16;
    const int mn   = lane & 15;             // row index for A, col index for B/C/D
    const int koff = (lane >> 4) << 1;      // 0 or 2

    const float* aRow = agg + (size_t)(row0 + mn) * DIM + koff;
    const float* hRow = h   + (size_t)(row0 + mn) * DIM + koff;
    const float* wRel = w_rel  + (size_t)koff * DIM + col0 + mn;
    const float* wRoo = w_root + (size_t)koff * DIM + col0 + mn;

    v8f acc = {};
    #pragma unroll 4
    for (int k = 0; k < DIM; k += 4) {
        // agg @ w_rel
        v2f a = *(const v2f*)(aRow + k);
        v2f b;
        b.x = wRel[(size_t)k * DIM];
        b.y = wRel[(size_t)(k + 1) * DIM];
        acc = __builtin_amdgcn_wmma_f32_16x16x4_f32(false, a, false, b,
                                                    (short)0, acc, false, false);
        // h @ w_root
        v2f a2 = *(const v2f*)(hRow + k);
        v2f b2;
        b2.x = wRoo[(size_t)k * DIM];
        b2.y = wRoo[(size_t)(k + 1) * DIM];
        acc = __builtin_amdgcn_wmma_f32_16x16x4_f32(false, a2, false, b2,
                                                    (short)0, acc, false, false);
    }

    const float bias = b_rel[col0 + mn];
    const int rbase = row0 + ((lane >> 4) << 3);   // +0 or +8
    #pragma unroll
    for (int g = 0; g < 8; ++g) {
        float v = acc[g] + bias;
        if (do_relu) v = fmaxf(v, 0.0f);
        out[(size_t)(rbase + g) * DIM + col0 + mn] = v;
    }
}

extern "C" void kernel_launch(void* const* d_in, const int* in_sizes, int n_in,
                              void* d_out, int out_size, void* d_ws, size_t ws_size,
                              hipStream_t stream) {
    (void)in_sizes; (void)n_in; (void)out_size; (void)ws_size;

    const float* x   = (const float*)d_in[0];
    const int*   ei  = (const int*)  d_in[1];   // [2, E] flat
    const float* ea  = (const float*)d_in[2];
    const float* w_rel[3]  = { (const float*)d_in[3], (const float*)d_in[6], (const float*)d_in[9]  };
    const float* b_rel[3]  = { (const float*)d_in[4], (const float*)d_in[7], (const float*)d_in[10] };
    const float* w_root[3] = { (const float*)d_in[5], (const float*)d_in[8], (const float*)d_in[11] };

    const int* src = ei;
    const int* dst = ei + N_EDGES;

    float* agg = (float*)d_ws;
    float* hA  = agg + (size_t)N_NODES * DIM;
    float* hB  = hA  + (size_t)N_NODES * DIM;
    float* outs[3] = { hA, hB, (float*)d_out };

    const int n4 = (N_NODES * DIM) / 4;
    const dim3 blk(256);
    const dim3 gZero((n4 + 255) / 256);
    const dim3 gScat(((size_t)N_EDGES * 32 + 255) / 256);
    const dim3 gGemm(STRIPES);

    const float* hin = x;
    for (int l = 0; l < 3; ++l) {
        zero_f4<<<gZero, blk, 0, stream>>>((float4*)agg, n4);
        scatter_edges<<<gScat, blk, 0, stream>>>(hin, src, dst, ea, agg);
        gemm_stripe<<<gGemm, blk, 0, stream>>>(agg, hin, w_rel[l], b_rel[l],
                                               w_root[l], outs[l], (l < 2) ? 1 : 0);
        hin = outs[l];
    }
}